// BinaryReflectanceGate_21809843929150
// MI455X (gfx1250) — compile-verified
//
#include <hip/hip_runtime.h>

// ---------------------------------------------------------------------------
// BinaryReflectanceGate for MI455X (gfx1250, wave32).
// Memory-bound (~134MB -> ~6us @ 23.3TB/s). fp32 WMMA MLP computed transposed
// (weights are the A operands of V_WMMA_F32_16X16X4_F32). C/D->B transpose via
// 4 unpredicated v_permlanex16 swaps feeding both sides of later selects.
// Loop control + segment detection are fully scalar: the wave index is forced
// uniform with readfirstlane, and since `batch` is sorted a tile is uniform
// iff its two endpoints match -> 2 SMEM probes replace 32 lane loads/tile.
// ---------------------------------------------------------------------------

typedef float v2f __attribute__((ext_vector_type(2)));
typedef float v8f __attribute__((ext_vector_type(8)));

#define NB 64
#define NH 16

__global__ __launch_bounds__(256) void init_seg_kernel(int* __restrict__ gseg) {
    int i = blockIdx.x * 256 + threadIdx.x;
    if (i < NB * NH) gseg[i] = 0;   // ReLU outputs >= 0, so 0 is a safe -inf
}

// Cross-half identity swap: lane l <- lane (l ^ 16). Pure VALU.
__device__ __forceinline__ float xswap16(float v) {
    return __int_as_float(__builtin_amdgcn_permlanex16(
        0, __float_as_int(v), 0x76543210, (int)0xfedcba98, true, false));
}

// A-layout (16x4 f32): lane<16: M=lane, V0=K0,V1=K1 ; lane>=16: V0=K2,V1=K3.
// B-layout (4x16 f32): V0: K0(lanes0-15)/K2(lanes16-31); V1: K1/K3.
// C/D-layout (16x16 f32): VGPR r: M=r (lanes0-15) / M=r+8 (lanes16-31).
__global__ __launch_bounds__(256) void mlp_segmax_kernel(
    const float* __restrict__ pos, const float* __restrict__ refl,
    const int* __restrict__ batch,
    const float* __restrict__ W1, const float* __restrict__ b1,
    const float* __restrict__ W2, const float* __restrict__ b2,
    int* __restrict__ gseg, int tiles_total)
{
    __shared__ int smax[NB * NH];
    for (int i = threadIdx.x; i < NB * NH; i += 256) smax[i] = 0;
    __syncthreads();

    const int  lane = threadIdx.x & 31;
    const int  n    = lane & 15;
    const bool lo   = lane < 16;
    const int  koff = lo ? 0 : 2;
    const int  hoff = lo ? 0 : 8;

    // Constant A operands (weights, transposed) and C operands (biases).
    v2f a1;
    a1.x = W1[(koff + 0) * NH + n];
    a1.y = W1[(koff + 1) * NH + n];
    v2f a2c[4];
#pragma unroll
    for (int c = 0; c < 4; ++c) {
        a2c[c].x = W2[(4 * c + koff + 0) * NH + n];
        a2c[c].y = W2[(4 * c + koff + 1) * NH + n];
    }
    v8f c1, c2;
#pragma unroll
    for (int r = 0; r < 8; ++r) { c1[r] = b1[r + hoff]; c2[r] = b2[r + hoff]; }

    // Lane-constant affine address bases/strides for the feature loads.
    const char* base_x = (const char*)(pos + koff);          // x|z, stride 12
    const char* base_y = lo ? (const char*)(pos + 1)         // y,   stride 12
                            : (const char*)refl;             // refl, stride 4
    const size_t st_y  = lo ? 12u : 4u;

    // Force the wave index uniform so loop control + batch probes go scalar.
    const int wave = __builtin_amdgcn_readfirstlane(
        blockIdx.x * (256 / 32) + (threadIdx.x >> 5));
    const int nwaves = gridDim.x * (256 / 32);

    // Register-resident running segment max (layer-2 ReLU folded in: the
    // 0-init dominates any negative pre-ReLU value).
    int cur = -1;
    v8f run;
#pragma unroll
    for (int r = 0; r < 8; ++r) run[r] = 0.0f;

    // Reduce run[] across the 16-lane halves and commit to LDS. Rare:
    // once per (wave, segment); <=63 segment changes grid-wide.
#define FLUSH_RUN()                                                          \
    do {                                                                     \
        _Pragma("unroll")                                                    \
        for (int r = 0; r < 8; ++r) {                                        \
            float v = run[r];                                                \
            _Pragma("unroll")                                                \
            for (int off = 1; off < 16; off <<= 1)                           \
                v = fmaxf(v, __shfl_xor(v, off, 32));                        \
            if (n == r)                                                      \
                atomicMax(&smax[cur * NH + r + hoff], __float_as_int(v));    \
            run[r] = 0.0f;                                                   \
        }                                                                    \
    } while (0)

    for (int tile = wave; tile < tiles_total; tile += nwaves) {
        const int base = tile * 16;        // uniform
        const int p    = base + n;

        // Scalar segment probe: sorted batch => endpoints equal <=> uniform.
        const int bid0  = batch[base];
        const int bid15 = batch[base + 15];

        // feat^T B operand: V0 = x | z, V1 = y | reflectance.
        v2f bv;
        bv.x = *(const float*)(base_x + (size_t)p * 12u);
        bv.y = *(const float*)(base_y + (size_t)p * st_y);

        // Layer 1 + ReLU
        v8f h = __builtin_amdgcn_wmma_f32_16x16x4_f32(
            false, a1, false, bv, (short)0, c1, false, false);
#pragma unroll
        for (int r = 0; r < 8; ++r) h[r] = fmaxf(h[r], 0.0f);

        // Build all four layer-2 B chunks. Each swap result (s*) carries
        // chunk c's high-lane data AND chunk c+2's low-lane data, so it is
        // consumed on both sides of the selects -> stays unpredicated.
        const float s0 = xswap16(lo ? h[2] : h[0]);
        const float s1 = xswap16(lo ? h[3] : h[1]);
        const float s4 = xswap16(lo ? h[6] : h[4]);
        const float s5 = xswap16(lo ? h[7] : h[5]);
        v2f b0, b1v, b2v, b3v;
        b0.x  = lo ? h[0] : s0;   b0.y  = lo ? h[1] : s1;   // rows 0..3
        b1v.x = lo ? h[4] : s4;   b1v.y = lo ? h[5] : s5;   // rows 4..7
        b2v.x = lo ? s0 : h[2];   b2v.y = lo ? s1 : h[3];   // rows 8..11
        b3v.x = lo ? s4 : h[6];   b3v.y = lo ? s5 : h[7];   // rows 12..15

        // Layer 2: 4 chained WMMAs (K = 16 split into 4x K=4).
        v8f acc = __builtin_amdgcn_wmma_f32_16x16x4_f32(
            false, a2c[0], false, b0, (short)0, c2, false, false);
        acc = __builtin_amdgcn_wmma_f32_16x16x4_f32(
            false, a2c[1], false, b1v, (short)0, acc, false, false);
        acc = __builtin_amdgcn_wmma_f32_16x16x4_f32(
            false, a2c[2], false, b2v, (short)0, acc, false, false);
        acc = __builtin_amdgcn_wmma_f32_16x16x4_f32(
            false, a2c[3], false, b3v, (short)0, acc, false, false);
        // NOTE: no explicit layer-2 ReLU -- folded into the max-with-0 below.

        if (bid0 == bid15) {          // uniform tile (scalar branch, common)
            if (bid0 != cur) {
                if (cur >= 0) FLUSH_RUN();
                cur = bid0;
            }
#pragma unroll
            for (int r = 0; r < 8; ++r) run[r] = fmaxf(run[r], acc[r]);
        } else {                      // segment-boundary tile (rare)
            // Per-lane batch ids needed only here. Negative float bits are
            // negative ints -> lose to the 0-init slots under int atomicMax,
            // so pre-ReLU values are safe.
            const int bid = batch[p];
#pragma unroll
            for (int r = 0; r < 8; ++r)
                atomicMax(&smax[bid * NH + r + hoff], __float_as_int(acc[r]));
        }
    }
    if (cur >= 0) FLUSH_RUN();
#undef FLUSH_RUN

    __syncthreads();
    for (int i = threadIdx.x; i < NB * NH; i += 256) {
        int v = smax[i];
        if (v != 0) atomicMax(&gseg[i], v);
    }
}

__global__ __launch_bounds__(64) void gate_kernel(
    const int* __restrict__ gseg, const float* __restrict__ gum,
    const float* __restrict__ Wg, const float* __restrict__ bg,
    float* __restrict__ gate)
{
    int t = threadIdx.x;
    if (t >= NB) return;
    float l0 = bg[0] + gum[2 * t + 0];
    float l1 = bg[1] + gum[2 * t + 1];
#pragma unroll
    for (int k = 0; k < NH; ++k) {
        float sf = __int_as_float(gseg[t * NH + k]);
        l0 += sf * Wg[2 * k + 0];
        l1 += sf * Wg[2 * k + 1];
    }
    // TAU == 1.0
    float m  = fmaxf(l0, l1);
    float e0 = __expf(l0 - m);
    float e1 = __expf(l1 - m);
    gate[t] = e1 / (e0 + e1);
}

__global__ __launch_bounds__(256) void scale_kernel(
    const int4* __restrict__ batch4, const float4* __restrict__ refl4,
    const float* __restrict__ gate, float4* __restrict__ out4, int n4)
{
    __shared__ float g[NB];
    if (threadIdx.x < NB) g[threadIdx.x] = gate[threadIdx.x];
    __syncthreads();
    int i = blockIdx.x * 256 + threadIdx.x;
    if (i >= n4) return;
    int4   b = batch4[i];
    float4 r = refl4[i];
    float4 o;
    o.x = g[b.x] * r.x;
    o.y = g[b.y] * r.y;
    o.z = g[b.z] * r.z;
    o.w = g[b.w] * r.w;
    out4[i] = o;
}

extern "C" void kernel_launch(void* const* d_in, const int* in_sizes, int n_in,
                              void* d_out, int out_size, void* d_ws, size_t ws_size,
                              hipStream_t stream) {
    const float* pos   = (const float*)d_in[0];
    const float* refl  = (const float*)d_in[1];
    const int*   batch = (const int*)d_in[2];
    const float* gum   = (const float*)d_in[3];
    const float* W1    = (const float*)d_in[4];
    const float* b1    = (const float*)d_in[5];
    const float* W2    = (const float*)d_in[6];
    const float* b2    = (const float*)d_in[7];
    const float* Wg    = (const float*)d_in[8];
    const float* bg    = (const float*)d_in[9];

    const int n = in_sizes[1];             // N points
    int*   gseg = (int*)d_ws;              // [64*16] segment-max (int bits)
    float* gate = (float*)d_ws + NB * NH;  // [64] gate weights

    init_seg_kernel<<<(NB * NH + 255) / 256, 256, 0, stream>>>(gseg);

    const int tiles = n / 16;
    mlp_segmax_kernel<<<1024, 256, 0, stream>>>(pos, refl, batch,
                                                W1, b1, W2, b2, gseg, tiles);

    gate_kernel<<<1, 64, 0, stream>>>(gseg, gum, Wg, bg, gate);

    const int n4 = n / 4;
    scale_kernel<<<(n4 + 255) / 256, 256, 0, stream>>>(
        (const int4*)batch, (const float4*)refl, gate, (float4*)d_out, n4);
}